// HDTimeCrystalBlock_82497731821688
// MI455X (gfx1250) — compile-verified
//
#include <hip/hip_runtime.h>
#include <math.h>

// Problem constants (from reference)
#define B_    4
#define L_    2048
#define DIN   512
#define HD_   4096
#define M_    16
#define TWO_PI 6.28318530717958647692f

typedef __attribute__((ext_vector_type(2))) float v2f;
typedef __attribute__((ext_vector_type(8))) float v8f;

#define WMMA_F32(a, b, c) \
  __builtin_amdgcn_wmma_f32_16x16x4_f32(false, (a), false, (b), (short)0, (c), false, false)

// Workspace layout (floats)
#define CA_OFF  0                         // [L_][M_]  coupled[m]*cos(2pi*(m+1)*l/L)
#define SA_OFF  (L_*M_)                   // [L_][M_]  coupled[m]*sin(...)
#define CE_OFF  (2*L_*M_)                 // [M_][HD_] cos(E[m][h])
#define SE_OFF  (2*L_*M_ + M_*HD_)        // [M_][HD_] sin(E[m][h])
#define MOD_OFF (2*L_*M_ + 2*M_*HD_)      // [L_][HD_] mod
// total ws floats = MOD_OFF + L_*HD_ = 196608 + 8388608  (~34.3 MB)

// ---------------------------------------------------------------------------
// Kernel 0: tiny trig tables (angle-addition factorization of mod)
// ---------------------------------------------------------------------------
__global__ void k_tables(const float* __restrict__ E,   // [M_][HD_]
                         const float* __restrict__ w,   // [M_]
                         const float* __restrict__ C,   // [M_][M_]
                         float* __restrict__ ws) {
  int idx = blockIdx.x * 256 + threadIdx.x;
  if (idx < L_ * M_) {
    int l = idx / M_, m = idx % M_;
    float cm = 0.f;                       // coupled[m] = C[m,:] . w
#pragma unroll
    for (int j = 0; j < M_; ++j) cm = fmaf(C[m * M_ + j], w[j], cm);
    float ang = TWO_PI * (float)(m + 1) * ((float)l / (float)L_);
    float s, c;
    sincosf(ang, &s, &c);
    ws[CA_OFF + idx] = cm * c;
    ws[SA_OFF + idx] = cm * s;
  } else {
    int j = idx - L_ * M_;
    if (j < M_ * HD_) {
      float e = E[j];
      float s, c;
      sincosf(e, &s, &c);
      ws[CE_OFF + j] = c;
      ws[SE_OFF + j] = s;
    }
  }
}

// ---------------------------------------------------------------------------
// Kernel 1: mod[l][h] = sum_m cA[l,m]*cE[m,h] - sA[l,m]*sE[m,h]  (pure FMA)
// ---------------------------------------------------------------------------
__global__ void k_mod(const float* __restrict__ ws, float* __restrict__ mod) {
  int idx = blockIdx.x * 256 + threadIdx.x;   // exact multiple of L_*HD_
  int l = idx >> 12;                          // / HD_
  int h = idx & (HD_ - 1);
  const float* cA = ws + CA_OFF + l * M_;
  const float* sA = ws + SA_OFF + l * M_;
  const float* cE = ws + CE_OFF + h;
  const float* sE = ws + SE_OFF + h;
  float acc = 0.f;
#pragma unroll
  for (int m = 0; m < M_; ++m) {
    acc = fmaf(cA[m],  cE[m * HD_], acc);
    acc = fmaf(-sA[m], sE[m * HD_], acc);
  }
  mod[idx] = acc;
}

// ---------------------------------------------------------------------------
// Kernel 2: fused  out[rows16] = ((x_tile @ W_in) * mod) @ W_out
// One workgroup (8 wave32s) owns 16 token rows; the full modulated 16x4096
// HD panel lives in LDS (~295KB of the 320KB WGP LDS).
// Register-blocked: 2 (stage1) / 4 (stage2) concurrent accumulator chains
// per wave share each A fragment -> fewer DS ops, independent wmma issues.
// ---------------------------------------------------------------------------
#define XS_STRIDE 516     // 512 + 4-float pad  -> 4-bank row skew
#define HD_STRIDE 4100    // 4096 + 4-float pad -> 4-bank row skew
#define SMEM_FLOATS (16 * XS_STRIDE + 16 * HD_STRIDE)

__global__ __launch_bounds__(256)
void k_fused(const float* __restrict__ x,     // [B*L][DIN]
             const float* __restrict__ Win,   // [DIN][HD_]
             const float* __restrict__ Wout,  // [HD_][DIN]
             const float* __restrict__ mod,   // [L_][HD_]
             float* __restrict__ out) {       // [B*L][DIN]
  extern __shared__ float smem[];
  float* xs = smem;                    // [16][XS_STRIDE]
  float* hd = smem + 16 * XS_STRIDE;   // [16][HD_STRIDE]

  const int row0 = blockIdx.x * 16;    // flat (b,l) row; tile never crosses b
  const int l0   = row0 & (L_ - 1);
  const int tid  = threadIdx.x;

  // --- stage 0: stage x tile (16 x 512) into LDS, float4 coalesced ---
  {
    const float4* xg = (const float4*)(x + (size_t)row0 * DIN);
    for (int i = tid; i < 16 * (DIN / 4); i += 256) {
      int r = i / (DIN / 4), c4 = i % (DIN / 4);
      float4 v = xg[r * (DIN / 4) + c4];
      *(float4*)&xs[r * XS_STRIDE + c4 * 4] = v;
    }
  }
  __syncthreads();

  const int lane = tid & 31;
  const int wave = tid >> 5;
  const int half = lane >> 4;   // K sub-phase of the WMMA fragment
  const int l16  = lane & 15;   // A-row / B-col / D-col within the 16x16 tile

  const float* xs_row = xs + l16 * XS_STRIDE;
  const float* hd_row = hd + l16 * HD_STRIDE;

  // --- stage 1: hd[16][4096] = xs @ Win, then * mod, into LDS ---
  // Each wave: 512 HD columns = 16 pairs of 16-wide tiles; the A fragment is
  // shared by both tiles of a pair (2 independent accumulator chains).
  for (int tp = 0; tp < 16; ++tp) {
    const int c0 = wave * 512 + tp * 32 + l16;
    v8f acc0 = {0.f, 0.f, 0.f, 0.f, 0.f, 0.f, 0.f, 0.f};
    v8f acc1 = {0.f, 0.f, 0.f, 0.f, 0.f, 0.f, 0.f, 0.f};
    const float* b0 = Win + c0;
    const float* b1 = Win + c0 + 16;
#pragma unroll 4
    for (int k0 = 0; k0 < DIN; k0 += 4) {
      const int ka = k0 + 2 * half;
      v2f a, v0, v1;
      a.x  = xs_row[ka];
      a.y  = xs_row[ka + 1];
      v0.x = b0[(size_t)ka * HD_];
      v0.y = b0[(size_t)(ka + 1) * HD_];
      v1.x = b1[(size_t)ka * HD_];
      v1.y = b1[(size_t)(ka + 1) * HD_];
      acc0 = WMMA_F32(a, v0, acc0);
      acc1 = WMMA_F32(a, v1, acc1);
    }
    // epilogue: Floquet modulation, stash in LDS
#pragma unroll
    for (int v = 0; v < 8; ++v) {
      const int r = v + 8 * half;
      const float* mrow = mod + (size_t)(l0 + r) * HD_;
      hd[r * HD_STRIDE + c0]      = acc0[v] * mrow[c0];
      hd[r * HD_STRIDE + c0 + 16] = acc1[v] * mrow[c0 + 16];
    }
  }
  __syncthreads();

  // --- stage 2: out[16][512] = hd @ Wout ---
  // Each wave: 64 output columns = 4 tiles processed concurrently; one A
  // fragment feeds 4 wmma (4 independent accumulator chains).
  {
    const int cb = wave * 64 + l16;
    v8f acc0 = {0.f, 0.f, 0.f, 0.f, 0.f, 0.f, 0.f, 0.f};
    v8f acc1 = {0.f, 0.f, 0.f, 0.f, 0.f, 0.f, 0.f, 0.f};
    v8f acc2 = {0.f, 0.f, 0.f, 0.f, 0.f, 0.f, 0.f, 0.f};
    v8f acc3 = {0.f, 0.f, 0.f, 0.f, 0.f, 0.f, 0.f, 0.f};
    const float* b0 = Wout + cb;
#pragma unroll 4
    for (int k0 = 0; k0 < HD_; k0 += 4) {
      const int ka = k0 + 2 * half;
      v2f a;
      a.x = hd_row[ka];
      a.y = hd_row[ka + 1];
      const float* br0 = b0 + (size_t)ka * DIN;
      const float* br1 = b0 + (size_t)(ka + 1) * DIN;
      v2f v0, v1, v2, v3;
      v0.x = br0[0];   v0.y = br1[0];
      v1.x = br0[16];  v1.y = br1[16];
      v2.x = br0[32];  v2.y = br1[32];
      v3.x = br0[48];  v3.y = br1[48];
      acc0 = WMMA_F32(a, v0, acc0);
      acc1 = WMMA_F32(a, v1, acc1);
      acc2 = WMMA_F32(a, v2, acc2);
      acc3 = WMMA_F32(a, v3, acc3);
    }
#pragma unroll
    for (int v = 0; v < 8; ++v) {
      const int r = v + 8 * half;
      float* orow = out + (size_t)(row0 + r) * DIN + cb;
      orow[0]  = acc0[v];
      orow[16] = acc1[v];
      orow[32] = acc2[v];
      orow[48] = acc3[v];
    }
  }
}

// ---------------------------------------------------------------------------
extern "C" void kernel_launch(void* const* d_in, const int* in_sizes, int n_in,
                              void* d_out, int out_size, void* d_ws, size_t ws_size,
                              hipStream_t stream) {
  (void)in_sizes; (void)n_in; (void)out_size; (void)ws_size;
  const float* x    = (const float*)d_in[0];  // [B,L,DIN]
  const float* Win  = (const float*)d_in[1];  // [DIN,HD]
  const float* Wout = (const float*)d_in[2];  // [HD,DIN]
  const float* E    = (const float*)d_in[3];  // [M,HD]
  const float* w    = (const float*)d_in[4];  // [M]
  const float* C    = (const float*)d_in[5];  // [M,M]
  float* out = (float*)d_out;
  float* ws  = (float*)d_ws;
  float* mod = ws + MOD_OFF;

  // K0: trig tables  (L*M + M*HD threads = 98304)
  k_tables<<<(L_ * M_ + M_ * HD_) / 256, 256, 0, stream>>>(E, w, C, ws);
  // K1: mod[L][HD]  (8.4M threads)
  k_mod<<<(L_ * HD_) / 256, 256, 0, stream>>>(ws, mod);
  // K2: fused double-GEMM, 512 row-tiles, ~295KB dynamic LDS each
  const size_t smem = SMEM_FLOATS * sizeof(float);
  (void)hipFuncSetAttribute((const void*)k_fused,
                            hipFuncAttributeMaxDynamicSharedMemorySize,
                            (int)smem);
  k_fused<<<(B_ * L_) / 16, 256, smem, stream>>>(x, Win, Wout, mod, out);
}